// GoldenSubjectDualStatisticsBatchNorm2d_5746666242191
// MI455X (gfx1250) — compile-verified
//
#include <hip/hip_runtime.h>
#include <math.h>

#define EPS_F 1e-5f
#define NDOM 8
#define B_ 64
#define C_ 256
#define H_ 32
#define W_ 128
#define HW_ 4096           // H_*W_
#define BC_ (B_ * C_)      // 16384

typedef __attribute__((ext_vector_type(2))) float v2f;
typedef __attribute__((ext_vector_type(8))) float v8f;

#if defined(__has_builtin)
#if __has_builtin(__builtin_amdgcn_wmma_f32_16x16x4_f32)
#define HAVE_WMMA_F32X4 1
#endif
#endif

// ---------------------------------------------------------------------------
// Kernel A: per-(b,c) spatial sum and sum-of-squares over HW=4096 contiguous
// floats. One wave32 per (b,c) slab; 32 iterations of fully-coalesced float4
// loads (512 B / wave / iter). Reduction is done on the matrix pipe:
// D = A x ones via V_WMMA_F32_16X16X4_F32 (full f32 accumulate, RNE).
// ---------------------------------------------------------------------------
__global__ __launch_bounds__(256) void bn_reduce_kernel(
    const float* __restrict__ x, float* __restrict__ s_out,
    float* __restrict__ sq_out) {
  const int wave = threadIdx.x >> 5;
  const int lane = threadIdx.x & 31;
  const int bc = blockIdx.x * 8 + wave;  // grid = BC_/8 blocks

  const float4* __restrict__ xv =
      (const float4*)(x + (size_t)bc * (size_t)HW_);

#ifdef HAVE_WMMA_F32X4
  v8f accS = {};
  v8f accQ = {};
  const v2f ones = {1.0f, 1.0f};
#pragma unroll 4
  for (int i = 0; i < 32; ++i) {
    float4 v = xv[i * 32 + lane];
    v2f a0 = {v.x, v.y};
    v2f a1 = {v.z, v.w};
    v2f q0 = {v.x * v.x, v.y * v.y};
    v2f q1 = {v.z * v.z, v.w * v.w};
    // D[m][n] = sum_k A[m][k] * 1.0 : every column holds the row sums.
    accS = __builtin_amdgcn_wmma_f32_16x16x4_f32(false, a0, false, ones,
                                                 (short)0, accS, false, false);
    accS = __builtin_amdgcn_wmma_f32_16x16x4_f32(false, a1, false, ones,
                                                 (short)0, accS, false, false);
    accQ = __builtin_amdgcn_wmma_f32_16x16x4_f32(false, q0, false, ones,
                                                 (short)0, accQ, false, false);
    accQ = __builtin_amdgcn_wmma_f32_16x16x4_f32(false, q1, false, ones,
                                                 (short)0, accQ, false, false);
  }
  // C/D layout: VGPR r, lanes 0-15 -> row r; lanes 16-31 -> row 8+r.
  // Per-lane partial covers rows 0-7 (lanes<16) or rows 8-15 (lanes>=16).
  float ps = accS[0] + accS[1] + accS[2] + accS[3] + accS[4] + accS[5] +
             accS[6] + accS[7];
  float pq = accQ[0] + accQ[1] + accQ[2] + accQ[3] + accQ[4] + accQ[5] +
             accQ[6] + accQ[7];
  ps += __shfl_xor(ps, 16, 32);
  pq += __shfl_xor(pq, 16, 32);
#else
  float ps = 0.0f, pq = 0.0f;
#pragma unroll 4
  for (int i = 0; i < 32; ++i) {
    float4 v = xv[i * 32 + lane];
    ps += v.x + v.y + v.z + v.w;
    pq += v.x * v.x + v.y * v.y + v.z * v.z + v.w * v.w;
  }
#pragma unroll
  for (int m = 16; m >= 1; m >>= 1) {
    ps += __shfl_xor(ps, m, 32);
    pq += __shfl_xor(pq, m, 32);
  }
#endif
  if (lane == 0) {
    s_out[bc] = ps;
    sq_out[bc] = pq;
  }
}

// ---------------------------------------------------------------------------
// Kernel B: deterministic segment-sum over B per (domain, channel); emits
// mean and invstd. 2048 outputs, trivial work (lives entirely in L2).
// ---------------------------------------------------------------------------
__global__ __launch_bounds__(256) void bn_stats_kernel(
    const float* __restrict__ s_in, const float* __restrict__ sq_in,
    const int* __restrict__ dom, float* __restrict__ mean_out,
    float* __restrict__ invstd_out) {
  const int e = blockIdx.x * 256 + threadIdx.x;
  if (e >= NDOM * C_) return;
  const int d = e / C_;
  const int c = e % C_;
  float S = 0.0f, Q = 0.0f, n = 0.0f;
#pragma unroll 8
  for (int b = 0; b < B_; ++b) {
    const float m = ((dom[b] - 1) == d) ? 1.0f : 0.0f;
    S = fmaf(m, s_in[b * C_ + c], S);
    Q = fmaf(m, sq_in[b * C_ + c], Q);
    n += m;
  }
  const float cnt = fmaxf(n * (float)HW_, 1.0f);
  const float mu = S / cnt;
  const float var = Q / cnt - mu * mu;
  mean_out[e] = mu;
  invstd_out[e] = 1.0f / sqrtf(var + EPS_F);
}

// ---------------------------------------------------------------------------
// Kernel C: streaming normalize. One block per (b,c) slab; per-block scalar
// stats folded to a single fma per element. 512 MB of traffic — the roofline
// term that dominates end-to-end time.
// ---------------------------------------------------------------------------
__global__ __launch_bounds__(256) void bn_norm_kernel(
    const float* __restrict__ x, const int* __restrict__ dom,
    const float* __restrict__ weight, const float* __restrict__ bias,
    const float* __restrict__ mean_in, const float* __restrict__ invstd_in,
    float* __restrict__ out) {
  const int bc = blockIdx.x;
  const int b = bc / C_;
  const int c = bc % C_;
  const int d = dom[b] - 1;
  const float mu = mean_in[d * C_ + c];
  const float istd = invstd_in[d * C_ + c];
  const float scale = istd * weight[c];
  const float shift = fmaf(-mu, scale, bias[c]);

  const float4* __restrict__ xv =
      (const float4*)(x + (size_t)bc * (size_t)HW_);
  float4* __restrict__ ov = (float4*)(out + (size_t)bc * (size_t)HW_);
#pragma unroll
  for (int i = 0; i < 4; ++i) {
    const int j = i * 256 + threadIdx.x;
    float4 v = xv[j];
    float4 r;
    r.x = fmaf(v.x, scale, shift);
    r.y = fmaf(v.y, scale, shift);
    r.z = fmaf(v.z, scale, shift);
    r.w = fmaf(v.w, scale, shift);
    ov[j] = r;
  }
}

// ---------------------------------------------------------------------------
// Launch. Inputs (setup_inputs order): x f32[64,256,32,128], weight f32[256],
// bias f32[256], domain_ids i32[64]. Output f32[64,256,32,128].
// Workspace layout (floats): s[16384] | sq[16384] | mean[2048] | invstd[2048]
// = 147456 bytes.
// ---------------------------------------------------------------------------
extern "C" void kernel_launch(void* const* d_in, const int* in_sizes, int n_in,
                              void* d_out, int out_size, void* d_ws,
                              size_t ws_size, hipStream_t stream) {
  (void)in_sizes; (void)n_in; (void)out_size; (void)ws_size;
  const float* x = (const float*)d_in[0];
  const float* weight = (const float*)d_in[1];
  const float* bias = (const float*)d_in[2];
  const int* dom = (const int*)d_in[3];
  float* out = (float*)d_out;

  float* ws = (float*)d_ws;
  float* s_buf = ws;                 // [BC_]
  float* sq_buf = ws + BC_;          // [BC_]
  float* mean_buf = ws + 2 * BC_;    // [NDOM*C_]
  float* invstd_buf = mean_buf + NDOM * C_;

  bn_reduce_kernel<<<BC_ / 8, 256, 0, stream>>>(x, s_buf, sq_buf);
  bn_stats_kernel<<<(NDOM * C_ + 255) / 256, 256, 0, stream>>>(
      s_buf, sq_buf, dom, mean_buf, invstd_buf);
  bn_norm_kernel<<<BC_, 256, 0, stream>>>(x, dom, weight, bias, mean_buf,
                                          invstd_buf, out);
}